// ColPali2Loss_8615704395980
// MI455X (gfx1250) — compile-verified
//
#include <hip/hip_runtime.h>
#include <hip/hip_bf16.h>

// ---------------------------------------------------------------------------
// ColPali2 loss on MI455X (gfx1250, wave32).
// Dominant work: sim[b,c,n,s] = q_multi[b] (32x128) @ d_multi[c]^T (128x1024)
// for 32x32 (b,c) pairs => ~8.6 GFLOP, 17 MB inputs (L2-resident on 192MB L2).
// Uses native fp32 tensor op V_WMMA_F32_16X16X4_F32 to match fp32 reference
// numerics; problem is latency-bound so fp32 WMMA is the right precision.
// ---------------------------------------------------------------------------

typedef __attribute__((ext_vector_type(2))) float v2f;
typedef __attribute__((ext_vector_type(8))) float v8f;

#define NB   32      // batch
#define NQ   32      // query vectors per sample
#define NDOC 1024    // doc vectors per sample
#define DIM  128     // embedding dim

#define ALPHA 0.5f
#define BETA  0.5f
#define TEMP  2.0f
#define EPS   1e-6f

__device__ __forceinline__ v8f wmma_f32_k4(v2f a, v2f b, v8f c) {
    // D = A(16x4 f32) * B(4x16 f32) + C(16x16 f32)
    return __builtin_amdgcn_wmma_f32_16x16x4_f32(
        /*neg_a=*/false, a, /*neg_b=*/false, b,
        /*c_mod=*/(short)0, c, /*reuse_a=*/false, /*reuse_b=*/false);
}

__device__ __forceinline__ float logit_eps(float x) {
    x = fminf(fmaxf(x, EPS), 1.0f - EPS);
    return logf(x) - log1pf(-x);
}

// ---------------------------------------------------------------------------
// Kernel 1: sv_scores (32x32) = q_single @ d_single^T, K=128.
// One WG, 4 waves, one 16x16 WMMA tile per wave.
// ---------------------------------------------------------------------------
__global__ __launch_bounds__(128) void sv_scores_kernel(
    const float* __restrict__ q, const float* __restrict__ d,
    float* __restrict__ sv) {
    const int lane = threadIdx.x & 31;
    const int w    = threadIdx.x >> 5;
    const int mt   = (w >> 1) * 16;   // row tile offset
    const int nt   = (w & 1) * 16;    // col tile offset
    const int l16  = lane & 15;
    const int hi   = lane >> 4;

    // A fragment: element (M=l16, K=k0+2*hi+j) in vgpr j. B mirrors it
    // because B = d^T (row-major d gives B column-major == same pattern).
    const float* qa = q + (mt + l16) * DIM + 2 * hi;
    const float* db = d + (nt + l16) * DIM + 2 * hi;

    v8f acc = {0.f, 0.f, 0.f, 0.f, 0.f, 0.f, 0.f, 0.f};
#pragma unroll
    for (int k0 = 0; k0 < DIM; k0 += 4) {
        v2f a = *(const v2f*)(qa + k0);
        v2f b = *(const v2f*)(db + k0);
        acc = wmma_f32_k4(a, b, acc);
    }
    // C/D layout: vgpr r, lane l -> M = r + 8*(l>>4), N = l&15
#pragma unroll
    for (int r = 0; r < 8; ++r) {
        const int row = mt + r + 8 * hi;
        const int col = nt + l16;
        sv[row * NB + col] = acc[r];
    }
}

// ---------------------------------------------------------------------------
// Kernel 2: mv_scores[b][c] = sum_n max_s  q_multi[b,n,:] . d_multi[c,s,:]
// Grid (32,32); 8 waves/WG. Each wave: 8 s-tiles x 2 n-tiles x 32 K-steps
// = 512 v_wmma_f32_16x16x4_f32. B fragment shared by both n-tiles.
// Per-lane running row-max across s, then shfl/LDS reductions.
// ---------------------------------------------------------------------------
__global__ __launch_bounds__(256) void mv_scores_kernel(
    const float* __restrict__ qm, const float* __restrict__ dm,
    float* __restrict__ mv) {
    const int b    = blockIdx.x;
    const int c    = blockIdx.y;
    const int lane = threadIdx.x & 31;
    const int w    = threadIdx.x >> 5;
    const int l16  = lane & 15;
    const int hi   = lane >> 4;

    __shared__ float partial[8][32];   // [wave][row n] partial max over s

    const float* qa0 = qm + ((size_t)b * NQ + l16) * DIM + 2 * hi;      // rows 0..15
    const float* qa1 = qa0 + 16 * DIM;                                   // rows 16..31
    const float* dbase = dm + (size_t)c * NDOC * DIM;

    float pmax0[8], pmax1[8];
#pragma unroll
    for (int r = 0; r < 8; ++r) { pmax0[r] = -3.0e38f; pmax1[r] = -3.0e38f; }

    for (int st = 0; st < 8; ++st) {
        const int s0 = (w * 8 + st) * 16;
        const float* db = dbase + (size_t)(s0 + l16) * DIM + 2 * hi;
        if (st < 7) __builtin_prefetch(db + 16 * DIM, 0, 1);  // next s-tile row

        v8f acc0 = {0.f, 0.f, 0.f, 0.f, 0.f, 0.f, 0.f, 0.f};
        v8f acc1 = {0.f, 0.f, 0.f, 0.f, 0.f, 0.f, 0.f, 0.f};
#pragma unroll
        for (int k0 = 0; k0 < DIM; k0 += 4) {
            v2f a0 = *(const v2f*)(qa0 + k0);
            v2f a1 = *(const v2f*)(qa1 + k0);
            v2f bb = *(const v2f*)(db + k0);
            acc0 = wmma_f32_k4(a0, bb, acc0);
            acc1 = wmma_f32_k4(a1, bb, acc1);
        }
#pragma unroll
        for (int r = 0; r < 8; ++r) {
            pmax0[r] = fmaxf(pmax0[r], acc0[r]);
            pmax1[r] = fmaxf(pmax1[r], acc1[r]);
        }
    }

    // Row max across the 16 columns of each half-wave (masks 1..8 stay
    // inside each 16-lane half on wave32).
#pragma unroll
    for (int r = 0; r < 8; ++r) {
        float m0 = pmax0[r], m1 = pmax1[r];
#pragma unroll
        for (int mask = 1; mask <= 8; mask <<= 1) {
            m0 = fmaxf(m0, __shfl_xor(m0, mask, 32));
            m1 = fmaxf(m1, __shfl_xor(m1, mask, 32));
        }
        pmax0[r] = m0; pmax1[r] = m1;
    }
    if (l16 == 0) {
        // lane 0 (hi=0): rows r / 16+r ; lane 16 (hi=1): rows 8+r / 24+r
#pragma unroll
        for (int r = 0; r < 8; ++r) {
            partial[w][hi * 8 + r]      = pmax0[r];
            partial[w][16 + hi * 8 + r] = pmax1[r];
        }
    }
    __syncthreads();

    if (threadIdx.x < 32) {
        const int row = threadIdx.x;
        float m = partial[0][row];
#pragma unroll
        for (int ww = 1; ww < 8; ++ww) m = fmaxf(m, partial[ww][row]);
        // sum over the 32 rows (MaxSim sum over n)
#pragma unroll
        for (int mask = 1; mask < 32; mask <<= 1) m += __shfl_xor(m, mask, 32);
        if (row == 0) mv[b * NB + c] = m;
    }
}

// ---------------------------------------------------------------------------
// Kernel 3: losses. One wave, lane b owns row b of sv/mv.
// ---------------------------------------------------------------------------
__global__ __launch_bounds__(32) void finalize_kernel(
    const float* __restrict__ sv, const float* __restrict__ mv,
    float* __restrict__ out) {
    const int b = threadIdx.x;
    const float* svr = sv + b * NB;
    const float* mvr = mv + b * NB;

    // ---- single-vector CE ----
    float mx = -3.0e38f;
    for (int j = 0; j < NB; ++j) mx = fmaxf(mx, svr[j]);
    float se = 0.f;
    for (int j = 0; j < NB; ++j) se += expf(svr[j] - mx);
    float svc = (mx + logf(se)) - svr[b];

    // ---- multi-vector softplus margin ----
    const float pos = mvr[b];
    float neg = -3.0e38f;
    for (int j = 0; j < NB; ++j) {
        float v = mvr[j] - (j == b ? 1.0e6f : 0.0f);
        neg = fmaxf(neg, v);
    }
    const float x  = neg - pos;
    float sp = (x > 0.f) ? (x + log1pf(expf(-x))) : log1pf(expf(x));

    // ---- distillation (torch kl_div, batchmean) ----
    const float student = logit_eps(pos) * (1.0f / TEMP);  // this lane = col b
    float drow = 0.f;
    for (int j = 0; j < NB; ++j) {
        const float stj = __shfl(student, j, 32);          // student[col j]
        const float t   = logit_eps(svr[j] * (1.0f / (float)NQ)) * (1.0f / TEMP);
        if (t > 0.f) drow += t * (logf(t) - stj);
    }

    // ---- reduce across 32 rows ----
#pragma unroll
    for (int mask = 1; mask < 32; mask <<= 1) {
        svc  += __shfl_xor(svc,  mask, 32);
        sp   += __shfl_xor(sp,   mask, 32);
        drow += __shfl_xor(drow, mask, 32);
    }
    if (b == 0) {
        const float sv_loss = svc * (1.0f / NB);
        const float mv_loss = sp  * (1.0f / NB);
        const float distill = (TEMP * TEMP) * drow * (1.0f / NB);
        out[0] = sv_loss;
        out[1] = mv_loss;
        out[2] = distill;
        out[3] = ALPHA * sv_loss + (1.0f - ALPHA) * mv_loss + BETA * distill;
    }
}

// ---------------------------------------------------------------------------
extern "C" void kernel_launch(void* const* d_in, const int* in_sizes, int n_in,
                              void* d_out, int out_size, void* d_ws, size_t ws_size,
                              hipStream_t stream) {
    const float* q_single = (const float*)d_in[0];  // (32,128)
    const float* d_single = (const float*)d_in[1];  // (32,128)
    const float* q_multi  = (const float*)d_in[2];  // (32,32,128)
    const float* d_multi  = (const float*)d_in[3];  // (32,1024,128)

    float* ws = (float*)d_ws;
    float* sv = ws;            // 32*32 floats, fully rewritten each call
    float* mv = ws + NB * NB;  // 32*32 floats, fully rewritten each call
    float* out = (float*)d_out;

    sv_scores_kernel<<<1, 128, 0, stream>>>(q_single, d_single, sv);
    mv_scores_kernel<<<dim3(NB, NB), 256, 0, stream>>>(q_multi, d_multi, mv);
    finalize_kernel<<<1, 32, 0, stream>>>(sv, mv, out);
}